// SpatialEncodeAgent_12146167513574
// MI455X (gfx1250) — compile-verified
//
#include <hip/hip_runtime.h>
#include <cstdint>

// Problem constants from the reference: S=100, C=64, cells per batch = S*S.
#define SS      10000        // S*S
#define CCH     64           // channels

typedef float v4f __attribute__((ext_vector_type(4)));

// ---------------------------------------------------------------------------
// Kernel 1: fill output with -inf (segment_max identity).
// ---------------------------------------------------------------------------
__global__ void se_init_neg_inf(float4* __restrict__ out, int n4) {
    const float ninf = -__builtin_huge_valf();
    float4 v = make_float4(ninf, ninf, ninf, ninf);
    int stride = gridDim.x * blockDim.x;
    for (int i = blockIdx.x * blockDim.x + threadIdx.x; i < n4; i += stride)
        out[i] = v;   // global_store_b128 (RT: keep output resident in L2)
}

// ---------------------------------------------------------------------------
// Kernel 2: scatter-max. 16 lanes per agent; each lane handles 4 channels
// via one non-temporal b128 load and 4 native CDNA5 float-max atomics
// (non-returning, device scope, immediate channel offsets).
// ---------------------------------------------------------------------------
__global__ void se_scatter_max(const float* __restrict__ enc,
                               const int*   __restrict__ coords,
                               float*       __restrict__ out,
                               int nquads) {
    int p = blockIdx.x * blockDim.x + threadIdx.x;
    if (p >= nquads) return;

    int agent = p >> 4;          // 16 quads per agent
    int quad  = p & 15;          // which group of 4 channels

    // Streaming data (read once, 256 MB): non-temporal so it does not evict
    // the L2-resident output the atomics are hammering.
    const v4f v = __builtin_nontemporal_load(
        (const v4f*)(enc + (size_t)agent * CCH + (size_t)(quad << 2)));

    int coord = __builtin_nontemporal_load(coords + agent);  // b*SS + y*S + x
    int b     = coord / SS;
    int cell  = coord - b * SS;          // y*S + x

    // out[b][c][cell], c = quad*4 .. quad*4+3 ; channel stride = SS floats.
    float* base = out + (size_t)b * (CCH * SS) + (size_t)(quad << 2) * SS + cell;

    // Native CDNA5 fp32 max atomics (no CAS loop), device scope so all WGPs
    // agree in L2. Output (82 MB) is L2-resident (192 MB), so these stay
    // on-chip. Non-returning -> STOREcnt only; s_endpgm waits implicitly.
    asm volatile("global_atomic_max_num_f32 %0, %1, off scope:SCOPE_DEV"
                 :: "v"(base), "v"(v.x) : "memory");
    asm volatile("global_atomic_max_num_f32 %0, %1, off offset:40000 scope:SCOPE_DEV"
                 :: "v"(base), "v"(v.y) : "memory");
    asm volatile("global_atomic_max_num_f32 %0, %1, off offset:80000 scope:SCOPE_DEV"
                 :: "v"(base), "v"(v.z) : "memory");
    asm volatile("global_atomic_max_num_f32 %0, %1, off offset:120000 scope:SCOPE_DEV"
                 :: "v"(base), "v"(v.w) : "memory");
}

// ---------------------------------------------------------------------------
// Kernel 3: empty cells still hold -inf -> replace with 0 (reference's
// where(occupied, seg_max, 0)). Occupied cells are finite (inputs are
// normal-distributed), so testing for -inf is exact.
// ---------------------------------------------------------------------------
__global__ void se_finalize(float4* __restrict__ out, int n4) {
    const float ninf = -__builtin_huge_valf();
    int stride = gridDim.x * blockDim.x;
    for (int i = blockIdx.x * blockDim.x + threadIdx.x; i < n4; i += stride) {
        float4 v = out[i];               // hits L2 (output resident)
        v.x = (v.x == ninf) ? 0.0f : v.x;
        v.y = (v.y == ninf) ? 0.0f : v.y;
        v.z = (v.z == ninf) ? 0.0f : v.z;
        v.w = (v.w == ninf) ? 0.0f : v.w;
        out[i] = v;
    }
}

// ---------------------------------------------------------------------------
// Launch: d_in[0]=batch_size(scalar), d_in[1]=agent_encodings, d_in[2]=coords.
// N = in_sizes[2]; out_size = B*C*S*S.
// ---------------------------------------------------------------------------
extern "C" void kernel_launch(void* const* d_in, const int* in_sizes, int n_in,
                              void* d_out, int out_size, void* d_ws, size_t ws_size,
                              hipStream_t stream) {
    const float* enc    = (const float*)d_in[1];
    const int*   coords = (const int*)  d_in[2];
    float*       out    = (float*)      d_out;

    const int N      = in_sizes[2];        // number of agents
    const int n4     = out_size >> 2;      // output float4 count (out_size % 4 == 0)
    const int nquads = N * 16;             // (agent, 4-channel-group) pairs

    const int BLK = 256;                   // 8 wave32 per block

    se_init_neg_inf<<<(n4 + BLK - 1) / BLK, BLK, 0, stream>>>((float4*)out, n4);
    se_scatter_max <<<(nquads + BLK - 1) / BLK, BLK, 0, stream>>>(enc, coords, out, nquads);
    se_finalize    <<<(n4 + BLK - 1) / BLK, BLK, 0, stream>>>((float4*)out, n4);
}